// RasterAttention_64484638982312
// MI455X (gfx1250) — compile-verified
//
#include <hip/hip_runtime.h>
#include <math.h>

typedef __attribute__((ext_vector_type(2))) float v2f;
typedef __attribute__((ext_vector_type(8))) float v8f;

#define HD 4
#define BD 8
#define CD 64
#define RD 64
#define LD 2048
#define EPSV 1e-5f

// ---------------------------------------------------------------------------
// Kernel 1: KV projection.  For each (h,b):  [key_w[h]; value_w[h]] (128x64)
// @ inputs[h,b] (64x2048)  ->  KV[h,b] (128x2048), rows 0..63 = K, 64..127 = V.
// WMMA f32 16x16x4, block = 8 waves, each wave owns one 16-row M strip and
// loops 4 N tiles of a 64-wide L chunk.
// ---------------------------------------------------------------------------
__global__ __launch_bounds__(256) void kv_gemm_kernel(
    const float* __restrict__ inp, const float* __restrict__ kw,
    const float* __restrict__ vw, float* __restrict__ KV) {
  int nch = blockIdx.x;                 // 32 chunks of 64 columns
  int b = blockIdx.y, h = blockIdx.z;
  __shared__ float Wl[128 * 68];        // 128x64 weights, stride 68
  __shared__ float Xl[64 * 72];         // 64x64 activations, stride 72 (2*72%64=16)
  int tid = threadIdx.x;
  for (int idx = tid; idx < 128 * 64; idx += 256) {
    int row = idx >> 6, c = idx & 63;
    float w = (row < 64) ? kw[(h * 64 + row) * 64 + c]
                         : vw[(h * 64 + (row - 64)) * 64 + c];
    Wl[row * 68 + c] = w;
  }
  int l0 = nch * 64;
  for (int idx = tid; idx < 64 * 64; idx += 256) {
    int c = idx >> 6, j = idx & 63;
    Xl[c * 72 + j] = inp[((size_t)((h * BD + b) * 64 + c)) * LD + l0 + j];
  }
  __syncthreads();
  int wave = tid >> 5, lane = tid & 31;
  int lr = lane & 15, lh = lane >> 4;
  int m0 = wave * 16;
  float* dst = KV + ((size_t)(h * BD + b) * 128) * LD;
  for (int nt = 0; nt < 4; ++nt) {
    int n0 = nt * 16;
    v8f acc = {};
    for (int k0 = 0; k0 < 64; k0 += 4) {
      int ka = k0 + 2 * lh;
      v2f a, bb;
      a.x = Wl[(m0 + lr) * 68 + ka];
      a.y = Wl[(m0 + lr) * 68 + ka + 1];
      bb.x = Xl[ka * 72 + n0 + lr];
      bb.y = Xl[(ka + 1) * 72 + n0 + lr];
      acc = __builtin_amdgcn_wmma_f32_16x16x4_f32(false, a, false, bb,
                                                  (short)0, acc, false, false);
    }
#pragma unroll
    for (int j = 0; j < 8; ++j) {
      int row = m0 + j + (lh ? 8 : 0);
      dst[(size_t)row * LD + l0 + n0 + lr] = acc[j];
    }
  }
}

// ---------------------------------------------------------------------------
// Kernel 2: banded scores.  Thread per (h,b,l).  s[d] = q(.)K(.,l+d-2) over r.
// Z[l] = (L - nvalid) + sum exp(s);  ES[h,b,d,l] = (exp(s_d)-1)/Z[l];
// Zinv[h,b,l] = 1/Z[l].
// ---------------------------------------------------------------------------
__global__ __launch_bounds__(256) void attn_scores_kernel(
    const float* __restrict__ hidden, const float* __restrict__ KV,
    float* __restrict__ Zinv, float* __restrict__ ES) {
  int gid = blockIdx.x * 256 + threadIdx.x;   // H*B*L exactly
  int l = gid & (LD - 1);
  int b = (gid >> 11) & 7;
  int h = gid >> 14;
  int hb = h * BD + b;
  const float* Kp = KV + (size_t)hb * 128 * LD;
  float s[5] = {0.f, 0.f, 0.f, 0.f, 0.f};
  for (int r = 0; r < RD; ++r) {
    float qv = hidden[(b * RD + r) * LD + l] * 0.125f;  // 1/sqrt(R)
    const float* Kr = Kp + (size_t)r * LD;
#pragma unroll
    for (int d = 0; d < 5; ++d) {
      int m = l + d - 2;
      if (m >= 0 && m < LD) s[d] += qv * Kr[m];
    }
  }
  float Z = 0.f, em1[5];
  int nv = 0;
#pragma unroll
  for (int d = 0; d < 5; ++d) {
    int m = l + d - 2;
    if (m >= 0 && m < LD) {
      float e = expf(s[d]);
      Z += e;
      em1[d] = e - 1.f;
      nv++;
    } else {
      em1[d] = 0.f;
    }
  }
  Z += (float)(LD - nv);                 // off-band entries contribute exp(0)=1
  float zi = 1.f / Z;
  Zinv[hb * LD + l] = zi;
#pragma unroll
  for (int d = 0; d < 5; ++d) ES[((hb * 5 + d) * LD) + l] = em1[d] * zi;
}

// ---------------------------------------------------------------------------
// Kernel 3: usum[h,b,c] = sum_l V[c,l] / Z[l]
// ---------------------------------------------------------------------------
__global__ __launch_bounds__(256) void usum_kernel(
    const float* __restrict__ KV, const float* __restrict__ Zinv,
    float* __restrict__ USUM) {
  int blk = blockIdx.x;                  // (h*8+b)*64 + c
  int c = blk & 63;
  int hb = blk >> 6;
  const float* Vrow = KV + ((size_t)hb * 128 + 64 + c) * LD;
  const float* Zp = Zinv + hb * LD;
  float acc = 0.f;
  for (int l = threadIdx.x; l < LD; l += 256) acc += Vrow[l] * Zp[l];
  __shared__ float red[256];
  red[threadIdx.x] = acc;
  __syncthreads();
  for (int off = 128; off > 0; off >>= 1) {
    if (threadIdx.x < off) red[threadIdx.x] += red[threadIdx.x + off];
    __syncthreads();
  }
  if (threadIdx.x == 0) USUM[blk] = red[0];
}

// ---------------------------------------------------------------------------
// Kernel 4: attention output + residual + head-concat.
// val[c,m] = usum[c] + sum_{d'=-2..2, l=m+d'} ES[h,b,d'+2,l] * V[c,l]
// X[b, h*64+c, m] = inputs[h,b,c,m] + val
// ---------------------------------------------------------------------------
__global__ __launch_bounds__(256) void attn_out_kernel(
    const float* __restrict__ inp, const float* __restrict__ KV,
    const float* __restrict__ ES, const float* __restrict__ USUM,
    float* __restrict__ X) {
  int gid = blockIdx.x * 256 + threadIdx.x;   // H*B*C*L exactly (2^22)
  int m = gid & (LD - 1);
  int c = (gid >> 11) & 63;
  int b = (gid >> 17) & 7;
  int h = gid >> 20;
  int hb = h * BD + b;
  const float* Vrow = KV + ((size_t)hb * 128 + 64 + c) * LD;
  float val = USUM[hb * 64 + c];
#pragma unroll
  for (int d = 0; d < 5; ++d) {
    int lpos = m + d - 2;
    if (lpos >= 0 && lpos < LD)
      val += ES[((hb * 5 + d) * LD) + lpos] * Vrow[lpos];
  }
  float e = inp[((size_t)(hb * 64 + c)) * LD + m] + val;
  X[((size_t)(b * 256 + h * 64 + c)) * LD + m] = e;
}

// ---------------------------------------------------------------------------
// Kernel 5: channel-attention gate (A=8 bottleneck), thread per (b,l).
// ---------------------------------------------------------------------------
__global__ __launch_bounds__(256) void channel_gate_kernel(
    const float* __restrict__ hidden, const float* __restrict__ w1,
    const float* __restrict__ b1, const float* __restrict__ w2,
    const float* __restrict__ b2, float* __restrict__ X) {
  int gid = blockIdx.x * 256 + threadIdx.x;   // B*L exactly
  int l = gid & (LD - 1);
  int b = gid >> 11;
  float a[8];
#pragma unroll
  for (int j = 0; j < 8; ++j) a[j] = b1[j];
  for (int ch = 0; ch < 256; ++ch) {
    float xv = X[((size_t)(b * 256 + ch)) * LD + l];
#pragma unroll
    for (int j = 0; j < 8; ++j) a[j] += w1[j * 320 + ch] * xv;
  }
  for (int r = 0; r < RD; ++r) {
    float hv = hidden[(b * RD + r) * LD + l];
#pragma unroll
    for (int j = 0; j < 8; ++j) a[j] += w1[j * 320 + 256 + r] * hv;
  }
#pragma unroll
  for (int j = 0; j < 8; ++j) a[j] = fmaxf(a[j], 0.f);
  for (int o = 0; o < 256; ++o) {
    float g = b2[o];
#pragma unroll
    for (int j = 0; j < 8; ++j) g += w2[o * 8 + j] * a[j];
    float gate = 1.f / (1.f + expf(-g));
    size_t xi = ((size_t)(b * 256 + o)) * LD + l;
    X[xi] = X[xi] * gate;
  }
}

// ---------------------------------------------------------------------------
// Kernel 6: 3-tap conv + BN + ReLU as 3 shifted WMMA GEMMs.
// out[b,o,l] = relu(bn(sum_{i<Cin,t<3} w[o,i,t] * in[b,i,l+t-1])), M=64.
// Block = (l chunk of 128, b); 8 waves, wave w owns N tile w, loops 4 M tiles.
// Input channels processed in chunks of 64 through a halo'd LDS tile.
// ---------------------------------------------------------------------------
__global__ __launch_bounds__(256) void conv_bn_relu_kernel(
    const float* __restrict__ in, const float* __restrict__ w,
    const float* __restrict__ bng, const float* __restrict__ bnb,
    const float* __restrict__ bnm, const float* __restrict__ bnv,
    float* __restrict__ out, int Cin) {
  int l0 = blockIdx.x * 128;
  int b = blockIdx.y;
  __shared__ float xs[64 * 136];   // 64 in-ch x 130 halo cols, stride 136
  __shared__ float wsh[64 * 68];   // 64 out-ch x 64 in-ch, stride 68
  int tid = threadIdx.x;
  int wave = tid >> 5, lane = tid & 31;
  int lr = lane & 15, lh = lane >> 4;
  v8f acc[4] = {{}, {}, {}, {}};
  int nKC = Cin >> 6;
  for (int kc = 0; kc < nKC; ++kc) {
    __syncthreads();                       // prior reads of xs/wsh complete
    for (int idx = tid; idx < 64 * 130; idx += 256) {
      int i = idx / 130, j = idx % 130;
      int lg = l0 - 1 + j;                 // SAME padding
      float v = 0.f;
      if (lg >= 0 && lg < LD)
        v = in[((size_t)(b * Cin + kc * 64 + i)) * LD + lg];
      xs[i * 136 + j] = v;
    }
    for (int t = 0; t < 3; ++t) {
      if (t > 0) __syncthreads();          // prior reads of wsh complete
      for (int idx = tid; idx < 64 * 64; idx += 256) {
        int o = idx >> 6, i = idx & 63;
        wsh[o * 68 + i] = w[(o * Cin + kc * 64 + i) * 3 + t];
      }
      __syncthreads();                     // xs + wsh visible
      int ncol = wave * 16 + lr + t;       // halo col = out col + t
      for (int m = 0; m < 4; ++m) {
        for (int k0 = 0; k0 < 64; k0 += 4) {
          int ka = k0 + 2 * lh;
          v2f a, bb;
          a.x = wsh[(m * 16 + lr) * 68 + ka];
          a.y = wsh[(m * 16 + lr) * 68 + ka + 1];
          bb.x = xs[ka * 136 + ncol];
          bb.y = xs[(ka + 1) * 136 + ncol];
          acc[m] = __builtin_amdgcn_wmma_f32_16x16x4_f32(
              false, a, false, bb, (short)0, acc[m], false, false);
        }
      }
    }
  }
  int lcol = l0 + wave * 16 + lr;
#pragma unroll
  for (int m = 0; m < 4; ++m) {
#pragma unroll
    for (int j = 0; j < 8; ++j) {
      int o = m * 16 + j + (lh ? 8 : 0);
      float inv = bng[o] * rsqrtf(bnv[o] + EPSV);
      float y = acc[m][j] * inv + (bnb[o] - bnm[o] * inv);
      out[((size_t)(b * 64 + o)) * LD + lcol] = fmaxf(y, 0.f);
    }
  }
}

// ---------------------------------------------------------------------------
extern "C" void kernel_launch(void* const* d_in, const int* in_sizes, int n_in,
                              void* d_out, int out_size, void* d_ws,
                              size_t ws_size, hipStream_t stream) {
  const float* inputs = (const float*)d_in[0];   // [H,B,C,L]
  const float* hidden = (const float*)d_in[1];   // [B,R,L]
  const float* key_w  = (const float*)d_in[2];   // [H,R,C]
  const float* value_w = (const float*)d_in[3];  // [H,C,C]
  const float* ca_w1 = (const float*)d_in[4];    // [A, HC+R]
  const float* ca_b1 = (const float*)d_in[5];
  const float* ca_w2 = (const float*)d_in[6];    // [HC, A]
  const float* ca_b2 = (const float*)d_in[7];
  const float* pw1   = (const float*)d_in[8];    // [R, HC, 3]
  const float* bn1_g = (const float*)d_in[9];
  const float* bn1_b = (const float*)d_in[10];
  const float* bn1_m = (const float*)d_in[11];
  const float* bn1_v = (const float*)d_in[12];
  const float* pw2   = (const float*)d_in[13];   // [R, R, 3]
  const float* bn2_g = (const float*)d_in[14];
  const float* bn2_b = (const float*)d_in[15];
  const float* bn2_m = (const float*)d_in[16];
  const float* bn2_v = (const float*)d_in[17];
  float* outp = (float*)d_out;                   // [B,R,L]

  // workspace layout (floats)
  float* ws = (float*)d_ws;
  float* KV   = ws;                                    // H*B*128*L = 8388608
  float* Zinv = KV + (size_t)HD * BD * 128 * LD;       // H*B*L     = 65536
  float* ES   = Zinv + (size_t)HD * BD * LD;           // H*B*5*L   = 327680
  float* USUM = ES + (size_t)HD * BD * 5 * LD;         // H*B*C     = 2048
  float* X    = USUM + (size_t)HD * BD * CD;           // B*256*L   = 4194304
  float* Y1   = X + (size_t)BD * 256 * LD;             // B*64*L    = 1048576

  // 1) KV projection GEMMs (WMMA f32)
  kv_gemm_kernel<<<dim3(LD / 64, BD, HD), 256, 0, stream>>>(inputs, key_w,
                                                            value_w, KV);
  // 2) banded scores / Z
  attn_scores_kernel<<<(HD * BD * LD) / 256, 256, 0, stream>>>(hidden, KV,
                                                               Zinv, ES);
  // 3) usum reduction
  usum_kernel<<<HD * BD * CD, 256, 0, stream>>>(KV, Zinv, USUM);
  // 4) attention output + residual + concat heads
  attn_out_kernel<<<(HD * BD * CD * LD) / 256, 256, 0, stream>>>(inputs, KV,
                                                                 ES, USUM, X);
  // 5) channel attention gate
  channel_gate_kernel<<<(BD * LD) / 256, 256, 0, stream>>>(hidden, ca_w1,
                                                           ca_b1, ca_w2,
                                                           ca_b2, X);
  // 6) conv1 (Cin=256) + BN1 + ReLU  (WMMA)
  conv_bn_relu_kernel<<<dim3(LD / 128, BD), 256, 0, stream>>>(
      X, pw1, bn1_g, bn1_b, bn1_m, bn1_v, Y1, 256);
  // 7) conv2 (Cin=64) + BN2 + ReLU -> out  (WMMA)
  conv_bn_relu_kernel<<<dim3(LD / 128, BD), 256, 0, stream>>>(
      Y1, pw2, bn2_g, bn2_b, bn2_m, bn2_v, outp, 64);
}